// DGKT_49022756717177
// MI455X (gfx1250) — compile-verified
//
#include <hip/hip_runtime.h>
#include <hip/hip_bf16.h>
#include <math.h>

// EGAT GNN forward for MI455X (gfx1250, wave32).
// Dense GEMMs: V_WMMA_F32_16X16X4_F32 (fp32-exact), B panel staged in LDS
// (ds_load_b64 fragments, conflict-free padded layout). Edge softmax and
// message aggregation use fp32 global atomics (L2-native on CDNA5).

typedef __attribute__((ext_vector_type(2))) float v2f;
typedef __attribute__((ext_vector_type(8))) float v8f;

#define NNODES 50000
#define NEDGES 320000
#define BATCH  4096
#define KMAX   256     // largest K used (agg GEMMs / final ui GEMM)

// ---------------------------------------------------------------- utilities
__device__ __forceinline__ void atomic_max_f32(float* addr, float val) {
    // classic signed/unsigned bit-pattern trick; works for all sign mixes
    if (val >= 0.0f) atomicMax((int*)addr, __float_as_int(val));
    else             atomicMin((unsigned int*)addr, __float_as_uint(val));
}

__global__ void fill_f32(float* __restrict__ p, float v, int n) {
    int i = blockIdx.x * blockDim.x + threadIdx.x;
    if (i < n) p[i] = v;
}

__global__ void fix_m_kernel(float* __restrict__ m, int n) {
    int i = blockIdx.x * blockDim.x + threadIdx.x;
    if (i < n && m[i] < -1e37f) m[i] = 0.0f;   // where(!isfinite(max), 0)
}

// ------------------------------------------------- WMMA fp32 GEMM, fused act
// C[M x N] = act(A[M x K] @ W[K x N] + bias); one wave -> one 16x16 tile,
// 8 waves/block share one 16-col W panel staged in LDS.
// act: 0 none, 1 elu, 2 leaky(0.01), 3 relu
__global__ void gemm_wmma_f32(const float* __restrict__ A, int lda,
                              const float* __restrict__ W, int ldw,
                              const float* __restrict__ bias,
                              float* __restrict__ C, int ldc,
                              int Mtiles, int K, int act) {
    // transposed, padded panel: bs[n*(K+2) + k]; pad=2 -> lane n uses banks
    // 2n,2n+1 for its ds_load_b64 -> conflict-free across 16 lanes.
    __shared__ float bs[16 * (KMAX + 2)];
    const int tid  = threadIdx.x;
    const int n0   = blockIdx.y * 16;
    const int kpad = K + 2;
    for (int i = tid; i < K * 16; i += 256) {
        int k = i >> 4, n = i & 15;
        bs[n * kpad + k] = W[(size_t)k * ldw + n0 + n];
    }
    __syncthreads();

    int lane = tid & 31;
    int wave = tid >> 5;
    int mt = blockIdx.x * 8 + wave;          // wave-uniform tile guard: EXEC
    if (mt < Mtiles) {                       // stays all-ones inside the wave
        int half = lane >> 4;                // 0: lanes 0-15, 1: lanes 16-31
        int l16  = lane & 15;
        int m0 = mt * 16;

        v8f c = {};
        const float* Arow = A  + (size_t)(m0 + l16) * lda;
        const float* Brow = bs + l16 * kpad;
        for (int k0 = 0; k0 < K; k0 += 4) {
            int ka = k0 + 2 * half;
            v2f a = *(const v2f*)(Arow + ka);   // A[l16][ka], A[l16][ka+1]
            v2f b = *(const v2f*)(Brow + ka);   // W[ka][n], W[ka+1][n] (LDS b64)
            c = __builtin_amdgcn_wmma_f32_16x16x4_f32(false, a, false, b,
                                                      (short)0, c, false, false);
        }
        float bb = bias ? bias[n0 + l16] : 0.0f;
#pragma unroll
        for (int r = 0; r < 8; ++r) {
            float v = c[r] + bb;
            if      (act == 1) v = (v > 0.0f) ? v : (expf(v) - 1.0f);
            else if (act == 2) v = (v > 0.0f) ? v : 0.01f * v;
            else if (act == 3) v = fmaxf(v, 0.0f);
            C[(size_t)(m0 + r + 8 * half) * ldc + n0 + l16] = v;
        }
    }
}

// ------------------------------------------------------------- edge kernels
// f = leaky0.2(xi[src] + xj[dst] + ew[e]) ; scores[e,h] = <f_h, attn_h>
// one wave per edge; lane covers 4 of 128 feats; head = lane>>3 (8 lanes/head)
__global__ void egat_edge_score(const float* __restrict__ xi,
                                const float* __restrict__ xj,
                                const float* __restrict__ ew,
                                const float* __restrict__ attn,   // (4,32)
                                const int* __restrict__ src,
                                const int* __restrict__ dst,
                                float* __restrict__ scores,       // (E,4)
                                float* __restrict__ m) {          // (N,4)
    int lane = threadIdx.x & 31;
    int e = blockIdx.x * 8 + (threadIdx.x >> 5);
    if (e >= NEDGES) return;
    int s = src[e], d = dst[e];
    int i0 = lane * 4;
    float4 fi = *(const float4*)(xi + (size_t)s * 128 + i0);
    float4 fj = *(const float4*)(xj + (size_t)d * 128 + i0);
    float4 fe = *(const float4*)(ew + (size_t)e * 128 + i0);
    int head = lane >> 3;
    float4 at = *(const float4*)(attn + head * 32 + (lane & 7) * 4);
    float p = 0.0f, v;
    v = fi.x + fj.x + fe.x; v = (v > 0.0f) ? v : 0.2f * v; p += v * at.x;
    v = fi.y + fj.y + fe.y; v = (v > 0.0f) ? v : 0.2f * v; p += v * at.y;
    v = fi.z + fj.z + fe.z; v = (v > 0.0f) ? v : 0.2f * v; p += v * at.z;
    v = fi.w + fj.w + fe.w; v = (v > 0.0f) ? v : 0.2f * v; p += v * at.w;
    p += __shfl_xor(p, 1, 32);
    p += __shfl_xor(p, 2, 32);
    p += __shfl_xor(p, 4, 32);
    if ((lane & 7) == 0) {
        scores[e * 4 + head] = p;
        atomic_max_f32(&m[(size_t)d * 4 + head], p);
    }
}

// ex = exp(score - m[dst]); denom[dst] += ex  (scores overwritten in place)
__global__ void egat_edge_exp(const int* __restrict__ dst,
                              float* __restrict__ scores,
                              const float* __restrict__ m,
                              float* __restrict__ denom) {
    int idx = blockIdx.x * blockDim.x + threadIdx.x;
    if (idx >= NEDGES * 4) return;
    int e = idx >> 2, h = idx & 3;
    int d = dst[e];
    float ex = expf(scores[idx] - m[(size_t)d * 4 + h]);
    scores[idx] = ex;
    atomicAdd(&denom[(size_t)d * 4 + h], ex);
}

// out[dst,h,:] += fn[src,h,:] * (ex/(denom+1e-9) * norm[e]); wave per edge,
// lane covers 8 consecutive of 256 feats (all within one head).
__global__ void egat_message(const float* __restrict__ fn,     // (N,256)
                             const float* __restrict__ ex,     // (E,4)
                             const float* __restrict__ denom,  // (N,4)
                             const float* __restrict__ norm,   // (E,)
                             const int* __restrict__ src,
                             const int* __restrict__ dst,
                             float* __restrict__ out) {        // (N,256)
    int lane = threadIdx.x & 31;
    int e = blockIdx.x * 8 + (threadIdx.x >> 5);
    if (e >= NEDGES) return;
    int s = src[e], d = dst[e];
    int head = lane >> 3;
    float a = ex[(size_t)e * 4 + head]
            / (denom[(size_t)d * 4 + head] + 1e-9f) * norm[e];
    int i0 = lane * 8;
    float4 f0 = *(const float4*)(fn + (size_t)s * 256 + i0);
    float4 f1 = *(const float4*)(fn + (size_t)s * 256 + i0 + 4);
    float* o = out + (size_t)d * 256 + i0;
    atomicAdd(o + 0, f0.x * a); atomicAdd(o + 1, f0.y * a);
    atomicAdd(o + 2, f0.z * a); atomicAdd(o + 3, f0.w * a);
    atomicAdd(o + 4, f1.x * a); atomicAdd(o + 5, f1.y * a);
    atomicAdd(o + 6, f1.z * a); atomicAdd(o + 7, f1.w * a);
}

// ------------------------------------------------------------ final layers
__global__ void gather_ui_kernel(const float* __restrict__ ui_states,
                                 const int* __restrict__ user_idx,
                                 const int* __restrict__ item_idx,
                                 float* __restrict__ ui_x) {   // (B,256)
    int idx = blockIdx.x * blockDim.x + threadIdx.x;
    if (idx >= BATCH * 256) return;
    int b = idx >> 8, c = idx & 255;
    ui_x[idx] = (c < 128) ? ui_states[(size_t)user_idx[b] * 128 + c]
                          : ui_states[(size_t)item_idx[b] * 128 + (c - 128)];
}

__global__ void gather_sg_kernel(const float* __restrict__ sg_states,
                                 const int* __restrict__ subg_idx,
                                 float* __restrict__ sg_x) {   // (B,128)
    int idx = blockIdx.x * blockDim.x + threadIdx.x;
    if (idx >= BATCH * 128) return;
    int b = idx >> 7, c = idx & 127;
    sg_x[idx] = sg_states[(size_t)subg_idx[b] * 128 + c];
}

__global__ void head_kernel(const float* __restrict__ hid_sg,
                            const float* __restrict__ w2_sg,
                            const float* __restrict__ b2_sg,
                            const float* __restrict__ hid_ui,
                            const float* __restrict__ w2_ui,
                            const float* __restrict__ b2_ui,
                            float* __restrict__ out) {
    int b = blockIdx.x * blockDim.x + threadIdx.x;
    if (b >= BATCH) return;
    float zs = b2_sg[0], zu = b2_ui[0];
    for (int i = 0; i < 128; ++i) {
        zs += hid_sg[(size_t)b * 128 + i] * w2_sg[i];
        zu += hid_ui[(size_t)b * 128 + i] * w2_ui[i];
    }
    out[b]         = 1.0f / (1.0f + expf(-zs));   // sg_out first (tuple order)
    out[BATCH + b] = 1.0f / (1.0f + expf(-zu));   // then ui_out
}

// ------------------------------------------------------------------- driver
extern "C" void kernel_launch(void* const* d_in, const int* in_sizes, int n_in,
                              void* d_out, int out_size, void* d_ws, size_t ws_size,
                              hipStream_t stream) {
    const float* x       = (const float*)d_in[0];
    const float* efeat   = (const float*)d_in[1];
    const float* efeat2  = (const float*)d_in[2];
    const float* emask   = (const float*)d_in[3];
    const float* emask2  = (const float*)d_in[4];
    const int*   src     = (const int*)d_in[5];
    const int*   dst     = (const int*)d_in[6];
    const int*   user_idx= (const int*)d_in[7];
    const int*   item_idx= (const int*)d_in[8];
    const int*   subg_idx= (const int*)d_in[9];
    const float* loc_wni = (const float*)d_in[10];
    const float* loc_wnj = (const float*)d_in[11];
    const float* loc_we  = (const float*)d_in[12];
    const float* loc_attn= (const float*)d_in[13];
    const float* loc_wnode=(const float*)d_in[14];
    const float* agg1_w  = (const float*)d_in[15];
    const float* agg1_b  = (const float*)d_in[16];
    const float* glob_wni= (const float*)d_in[17];
    const float* glob_wnj= (const float*)d_in[18];
    const float* glob_we = (const float*)d_in[19];
    const float* glob_attn=(const float*)d_in[20];
    const float* glob_wnode=(const float*)d_in[21];
    const float* agg2_w  = (const float*)d_in[22];
    const float* agg2_b  = (const float*)d_in[23];
    const float* w1_ui   = (const float*)d_in[24];
    const float* b1_ui   = (const float*)d_in[25];
    const float* w1_sg   = (const float*)d_in[26];
    const float* b1_sg   = (const float*)d_in[27];
    const float* w2_ui   = (const float*)d_in[28];
    const float* b2_ui   = (const float*)d_in[29];
    const float* w2_sg   = (const float*)d_in[30];
    const float* b2_sg   = (const float*)d_in[31];

    float* ws = (float*)d_ws;
    size_t off = 0;
    auto wsalloc = [&](size_t n) { float* p = ws + off; off += n; return p; };
    float* tmp_ni   = wsalloc((size_t)NNODES * 128);
    float* tmp_nj   = wsalloc((size_t)NNODES * 128);
    float* fn       = wsalloc((size_t)NNODES * 256);
    float* ew       = wsalloc((size_t)NEDGES * 128);
    float* scores   = wsalloc((size_t)NEDGES * 4);
    float* m        = wsalloc((size_t)NNODES * 4);
    float* denom    = wsalloc((size_t)NNODES * 4);
    float* out_agg  = wsalloc((size_t)NNODES * 256);
    float* ui_states= wsalloc((size_t)NNODES * 128);
    float* sg_states= wsalloc((size_t)NNODES * 128);
    float* ui_x     = wsalloc((size_t)BATCH * 256);
    float* sg_x     = wsalloc((size_t)BATCH * 128);
    float* hid_ui   = wsalloc((size_t)BATCH * 128);
    float* hid_sg   = wsalloc((size_t)BATCH * 128);

    auto gemm = [&](const float* A, int lda, const float* W, int Ncols,
                    const float* bias, float* C, int ldc, int M, int K, int act) {
        int Mtiles = M / 16;
        dim3 grid((Mtiles + 7) / 8, Ncols / 16);
        gemm_wmma_f32<<<grid, 256, 0, stream>>>(A, lda, W, Ncols, bias,
                                                C, ldc, Mtiles, K, act);
    };
    auto fill = [&](float* p, float v, int n) {
        fill_f32<<<(n + 255) / 256, 256, 0, stream>>>(p, v, n);
    };

    const int EBLK = (NEDGES + 7) / 8;  // wave-per-edge kernels, 8 waves/block

    auto egat = [&](const float* h, int hld, const float* ef, int efK,
                    const float* wni, const float* wnj, const float* we,
                    const float* attn, const float* wnode, const float* norm) {
        gemm(h,  hld, wni,   128, nullptr, tmp_ni, 128, NNODES, 64,  0);
        gemm(h,  hld, wnj,   128, nullptr, tmp_nj, 128, NNODES, 64,  0);
        gemm(ef, efK, we,    128, nullptr, ew,     128, NEDGES, efK, 0);
        gemm(h,  hld, wnode, 256, nullptr, fn,     256, NNODES, 64,  0);
        fill(m, -3.0e38f, NNODES * 4);
        fill(denom, 0.0f, NNODES * 4);
        egat_edge_score<<<EBLK, 256, 0, stream>>>(tmp_ni, tmp_nj, ew, attn,
                                                  src, dst, scores, m);
        fix_m_kernel<<<(NNODES * 4 + 255) / 256, 256, 0, stream>>>(m, NNODES * 4);
        egat_edge_exp<<<(NEDGES * 4 + 255) / 256, 256, 0, stream>>>(dst, scores,
                                                                    m, denom);
        fill(out_agg, 0.0f, NNODES * 256);
        egat_message<<<EBLK, 256, 0, stream>>>(fn, scores, denom, norm,
                                               src, dst, out_agg);
    };

    const float* h = x;
    int hld = 64;
    for (int l = 0; l < 2; ++l) {
        // local EGAT + agg1 (elu) -> ui_states[:, l*64 : l*64+64]
        egat(h, hld, efeat, 16,
             loc_wni + (size_t)l * 64 * 128, loc_wnj + (size_t)l * 64 * 128,
             loc_we + (size_t)l * 16 * 128,  loc_attn + (size_t)l * 4 * 32,
             loc_wnode + (size_t)l * 64 * 256, emask);
        gemm(out_agg, 256, agg1_w + (size_t)l * 256 * 64, 64,
             agg1_b + (size_t)l * 64, ui_states + l * 64, 128, NNODES, 256, 1);
        h = ui_states + l * 64; hld = 128;

        // global EGAT + agg2 (leaky 0.01) -> sg_states[:, l*64 : l*64+64]
        egat(h, hld, efeat2, 64,
             glob_wni + (size_t)l * 64 * 128, glob_wnj + (size_t)l * 64 * 128,
             glob_we + (size_t)l * 64 * 128,  glob_attn + (size_t)l * 4 * 32,
             glob_wnode + (size_t)l * 64 * 256, emask2);
        gemm(out_agg, 256, agg2_w + (size_t)l * 256 * 64, 64,
             agg2_b + (size_t)l * 64, sg_states + l * 64, 128, NNODES, 256, 2);
        h = sg_states + l * 64; hld = 128;
    }

    gather_ui_kernel<<<(BATCH * 256 + 255) / 256, 256, 0, stream>>>(
        ui_states, user_idx, item_idx, ui_x);
    gemm(ui_x, 256, w1_ui, 128, b1_ui, hid_ui, 128, BATCH, 256, 3);
    gather_sg_kernel<<<(BATCH * 128 + 255) / 256, 256, 0, stream>>>(
        sg_states, subg_idx, sg_x);
    gemm(sg_x, 128, w1_sg, 128, b1_sg, hid_sg, 128, BATCH, 128, 3);
    head_kernel<<<(BATCH + 255) / 256, 256, 0, stream>>>(
        hid_sg, w2_sg, b2_sg, hid_ui, w2_ui, b2_ui, (float*)d_out);
}